// GNN_node_30279519437416
// MI455X (gfx1250) — compile-verified
//
#include <hip/hip_runtime.h>
#include <hip/hip_bf16.h>

// ---------------------------------------------------------------------------
// CDNA5 (gfx1250) GNN pipeline.
// GEMMs use V_WMMA_F32_16X16X4_F32 (fp32 matrix pipe, wave32), register-blocked
// 32x64 per wave (2x4 tiles, 8 accumulators) for A-reuse and WMMA density.
// ---------------------------------------------------------------------------

typedef __attribute__((ext_vector_type(2))) float v2f;
typedef __attribute__((ext_vector_type(8))) float v8f;

#define EMB   128
#define LEAKY 0.1f

static __device__ __forceinline__ float leaky1(float v) {
    return v >= 0.0f ? v : LEAKY * v;
}

// ---------------------------------------------------------------------------
// fp32 WMMA GEMM:  C[M,Ncol] = act(A[M,K] @ B[K,Ncol] + bias[Ncol])
// One wave per 32x64 output block (2 M-tiles x 4 N-tiles); 8 waves per block.
// Requires M % 32 == 0, Ncol % 64 == 0, K % 4 == 0 (all hold here).
// Fragment layouts per CDNA5 ISA 7.12.2 (f32 16x16x4):
//   A: lane<16 -> M=lane,   {K+0,K+1}; lane>=16 -> M=lane-16, {K+2,K+3}
//   B: lane<16 -> N=lane,   rows {K+0,K+1}; lane>=16 -> N=lane-16, {K+2,K+3}
//   C: vgpr v, lane<16 -> M=v, N=lane; lane>=16 -> M=v+8, N=lane-16
// ---------------------------------------------------------------------------
__global__ __launch_bounds__(256) void wmma_gemm_f32(
    const float* __restrict__ A, const float* __restrict__ B,
    const float* __restrict__ bias, float* __restrict__ C,
    int M, int Ncol, int K, int act)
{
    const int nbN = Ncol >> 6;                       // 64-wide column blocks
    const int nbM = M >> 5;                          // 32-high row blocks
    const int blk = blockIdx.x * 8 + (threadIdx.x >> 5);
    if (blk >= nbM * nbN) return;

    const int bm   = blk / nbN;
    const int bn   = blk - bm * nbN;
    const int lane = threadIdx.x & 31;
    const int hb   = lane >> 4;                      // lane half (K sub-pair)
    const int l16  = lane & 15;

    const int row0 = bm << 5;
    const int col0 = bn << 6;

    v8f acc[2][4] = {};

    const float* aBase0 = A + (size_t)(row0 + l16) * K + 2 * hb;
    const float* aBase1 = aBase0 + (size_t)16 * K;
    const float* bBase  = B + (size_t)(2 * hb) * Ncol + col0 + l16;

    for (int k = 0; k < K; k += 4) {
        v2f a0, a1;
        a0.x = aBase0[k];
        a0.y = aBase0[k + 1];
        a1.x = aBase1[k];
        a1.y = aBase1[k + 1];
        const float* bp = bBase + (size_t)k * Ncol;
        v2f bf[4];
#pragma unroll
        for (int j = 0; j < 4; ++j) {
            bf[j].x = bp[16 * j];
            bf[j].y = bp[16 * j + Ncol];
        }
#pragma unroll
        for (int j = 0; j < 4; ++j) {
            acc[0][j] = __builtin_amdgcn_wmma_f32_16x16x4_f32(
                false, a0, false, bf[j], (short)0, acc[0][j], false, false);
            acc[1][j] = __builtin_amdgcn_wmma_f32_16x16x4_f32(
                false, a1, false, bf[j], (short)0, acc[1][j], false, false);
        }
    }

#pragma unroll
    for (int i = 0; i < 2; ++i) {
#pragma unroll
        for (int j = 0; j < 4; ++j) {
            const int n  = col0 + 16 * j + l16;
            const float bv = bias ? bias[n] : 0.0f;
            float* cp = C + (size_t)(row0 + 16 * i + 8 * hb) * Ncol + n;
#pragma unroll
            for (int v = 0; v < 8; ++v) {
                float val = acc[i][j][v] + bv;
                if (act) val = leaky1(val);
                cp[(size_t)v * Ncol] = val;
            }
        }
    }
}

// ---------------------------------------------------------------------------
// Degree count: degc[row] += 1 per edge (deg = degc + 1 downstream).
// ---------------------------------------------------------------------------
__global__ void deg_count_kernel(const int* __restrict__ ei,
                                 float* __restrict__ degc, int E)
{
    int t = blockIdx.x * blockDim.x + threadIdx.x;
    if (t < E) atomicAdd(&degc[ei[t]], 1.0f);
}

// ---------------------------------------------------------------------------
// Edge scatter: agg[col] += norm * relu(xl[row]); one thread per (edge, 4 cols)
// ---------------------------------------------------------------------------
__global__ void scatter_kernel(const float* __restrict__ xl,
                               const int* __restrict__ ei,
                               const float* __restrict__ degc,
                               float* __restrict__ agg, int E)
{
    long t = (long)blockIdx.x * blockDim.x + threadIdx.x;
    if (t >= (long)E * 32) return;
    const int e  = (int)(t >> 5);
    const int c4 = (int)(t & 31) << 2;
    const int row = ei[e];
    const int col = ei[E + e];
    const float norm = rsqrtf(degc[row] + 1.0f) * rsqrtf(degc[col] + 1.0f);
    const float4 xv = *(const float4*)(xl + (size_t)row * EMB + c4);
    float* ap = agg + (size_t)col * EMB + c4;
    atomicAdd(ap + 0, fmaxf(xv.x, 0.0f) * norm);
    atomicAdd(ap + 1, fmaxf(xv.y, 0.0f) * norm);
    atomicAdd(ap + 2, fmaxf(xv.z, 0.0f) * norm);
    atomicAdd(ap + 3, fmaxf(xv.w, 0.0f) * norm);
}

// ---------------------------------------------------------------------------
// Self term: agg[i][c] += relu(xl[i][c] + root[c]) / (degc[i] + 1)
// ---------------------------------------------------------------------------
__global__ void self_combine_kernel(const float* __restrict__ xl,
                                    const float* __restrict__ root,
                                    const float* __restrict__ degc,
                                    float* __restrict__ agg, int Nn)
{
    long t = (long)blockIdx.x * blockDim.x + threadIdx.x;
    if (t >= (long)Nn * EMB) return;
    const int i = (int)(t >> 7);
    const int c = (int)(t & (EMB - 1));
    const float s = fmaxf(xl[t] + root[c], 0.0f);
    agg[t] += s / (degc[i] + 1.0f);
}

// ---------------------------------------------------------------------------
// LayerNorm + leaky + residual; one wave (32 lanes) per 128-wide row.
// h[i] = leaky(LN(hc[i])*g + b) + h[i]; rows < NI also written to out slab l+1
// ---------------------------------------------------------------------------
__global__ __launch_bounds__(256) void ln_leaky_res_kernel(
    const float* __restrict__ hc, const float* __restrict__ g,
    const float* __restrict__ b, float* __restrict__ h,
    float* __restrict__ out, int Nn, int NI, int l)
{
    const int row = blockIdx.x * 8 + (threadIdx.x >> 5);
    if (row >= Nn) return;
    const int lane = threadIdx.x & 31;
    const float* hr = hc + (size_t)row * EMB;

    float v[4];
    float sum = 0.0f;
#pragma unroll
    for (int j = 0; j < 4; ++j) {
        v[j] = hr[lane + 32 * j];
        sum += v[j];
    }
#pragma unroll
    for (int off = 16; off > 0; off >>= 1) sum += __shfl_xor(sum, off, 32);
    const float mu = sum * (1.0f / EMB);

    float vs = 0.0f;
#pragma unroll
    for (int j = 0; j < 4; ++j) {
        const float d = v[j] - mu;
        vs += d * d;
    }
#pragma unroll
    for (int off = 16; off > 0; off >>= 1) vs += __shfl_xor(vs, off, 32);
    const float rs = rsqrtf(vs * (1.0f / EMB) + 1e-5f);

#pragma unroll
    for (int j = 0; j < 4; ++j) {
        const int c = lane + 32 * j;
        float y = (v[j] - mu) * rs * g[c] + b[c];
        y = leaky1(y);
        const float hn = y + h[(size_t)row * EMB + c];
        h[(size_t)row * EMB + c] = hn;
        if (row < NI) out[(size_t)row * (4 * EMB) + (size_t)(l + 1) * EMB + c] = hn;
    }
}

// ---------------------------------------------------------------------------
// Copy h0 (rows < NI) into output slab 0.
// ---------------------------------------------------------------------------
__global__ void copy_h0_kernel(const float* __restrict__ h,
                               float* __restrict__ out, int NI)
{
    long t = (long)blockIdx.x * blockDim.x + threadIdx.x;
    if (t >= (long)NI * EMB) return;
    const int i = (int)(t >> 7);
    const int c = (int)(t & (EMB - 1));
    out[(size_t)i * (4 * EMB) + c] = h[t];
}

static inline int cdiv(long a, long b) { return (int)((a + b - 1) / b); }

extern "C" void kernel_launch(void* const* d_in, const int* in_sizes, int n_in,
                              void* d_out, int out_size, void* d_ws, size_t ws_size,
                              hipStream_t stream)
{
    const float* x          = (const float*)d_in[0];
    const float* enc_w1     = (const float*)d_in[1];
    const float* enc_b1     = (const float*)d_in[2];
    const float* enc_w2     = (const float*)d_in[3];
    const float* enc_b2     = (const float*)d_in[4];
    const float* conv_w     = (const float*)d_in[5];
    const float* conv_b     = (const float*)d_in[6];
    const float* conv_root  = (const float*)d_in[7];
    const float* reconv_w   = (const float*)d_in[8];
    const float* reconv_b   = (const float*)d_in[9];
    const float* reconv_root= (const float*)d_in[10];
    const float* ln_g       = (const float*)d_in[11];
    const float* ln_b       = (const float*)d_in[12];
    const int*   e_nn       = (const int*)d_in[13];
    const int*   e_rn       = (const int*)d_in[14];

    const int N  = in_sizes[0] / 32;        // 200000
    const int E1 = in_sizes[13] / 2;        // 600000
    const int E2 = in_sizes[14] / 2;        // 600000
    const int NI = out_size / (4 * EMB);    // 120000

    float* out   = (float*)d_out;
    float* h     = (float*)d_ws;
    float* tmp1  = h    + (size_t)N * EMB;  // tmp1..tmp2 contiguous => [N,256]
    float* tmp2  = tmp1 + (size_t)N * EMB;
    float* degnn = tmp2 + (size_t)N * EMB;
    float* degrn = degnn + N;

    const size_t featBytes = (size_t)N * EMB * sizeof(float);

    // ---- degrees (recomputed every call: deterministic) ----
    hipMemsetAsync(degnn, 0, (size_t)N * sizeof(float), stream);
    hipMemsetAsync(degrn, 0, (size_t)N * sizeof(float), stream);
    deg_count_kernel<<<cdiv(E1, 256), 256, 0, stream>>>(e_nn, degnn, E1);
    deg_count_kernel<<<cdiv(E2, 256), 256, 0, stream>>>(e_rn, degrn, E2);

    // ---- node encoder: x[N,32] -> leaky -> [N,256] -> leaky -> h[N,128] ----
    {
        const int blks1 = (N / 32) * (256 / 64);
        wmma_gemm_f32<<<cdiv(blks1, 8), 256, 0, stream>>>(
            x, enc_w1, enc_b1, tmp1 /* t spans tmp1+tmp2 */, N, 256, 32, 1);
        const int blks2 = (N / 32) * (EMB / 64);
        wmma_gemm_f32<<<cdiv(blks2, 8), 256, 0, stream>>>(
            tmp1, enc_w2, enc_b2, h, N, EMB, 256, 1);
    }
    copy_h0_kernel<<<cdiv((long)NI * EMB, 256), 256, 0, stream>>>(h, out, NI);

    const int gblks = (N / 32) * (EMB / 64);
    for (int l = 0; l < 3; ++l) {
        const float* Wc = conv_w + (size_t)l * EMB * EMB;
        const float* Wr = reconv_w + (size_t)l * EMB * EMB;

        // GCN #1 (edge_index_node_net)
        wmma_gemm_f32<<<cdiv(gblks, 8), 256, 0, stream>>>(
            h, Wc, conv_b + l * EMB, tmp1, N, EMB, EMB, 0);
        hipMemsetAsync(tmp2, 0, featBytes, stream);
        scatter_kernel<<<cdiv((long)E1 * 32, 256), 256, 0, stream>>>(
            tmp1, e_nn, degnn, tmp2, E1);
        self_combine_kernel<<<cdiv((long)N * EMB, 256), 256, 0, stream>>>(
            tmp1, conv_root + l * EMB, degnn, tmp2, N);

        // GCN #2 (edge_index_net_node); input hc is in tmp2
        wmma_gemm_f32<<<cdiv(gblks, 8), 256, 0, stream>>>(
            tmp2, Wr, reconv_b + l * EMB, tmp1, N, EMB, EMB, 0);
        hipMemsetAsync(tmp2, 0, featBytes, stream);
        scatter_kernel<<<cdiv((long)E2 * 32, 256), 256, 0, stream>>>(
            tmp1, e_rn, degrn, tmp2, E2);
        self_combine_kernel<<<cdiv((long)N * EMB, 256), 256, 0, stream>>>(
            tmp1, reconv_root + l * EMB, degrn, tmp2, N);

        // LN + leaky + residual; h updated in place, slab l+1 of out written
        ln_leaky_res_kernel<<<cdiv(N, 8), 256, 0, stream>>>(
            tmp2, ln_g + l * EMB, ln_b + l * EMB, h, out, N, NI, l);
    }
}